// NBodyCGENN_73031623901583
// MI455X (gfx1250) — compile-verified
//
// CDNA5 / gfx1250 implementation of the NBody CGENN forward pass.
//
// Design notes (MI455X reasoning):
//  * ~19 GFLOP fp32 per forward, dominated by per-grade 28x28 channel-mixing
//    GEMMs over 70,656 edges -> mapped onto V_WMMA_F32_16X16X4_F32 (full-fp32
//    matrix pipe, K-step 4 fits K=28/56 exactly, no precision repack).
//  * Workgroup = 8 wave32; wave i owns multivector component i; the 8 waves
//    exchange components through LDS so mv_silu / norm_layer (which mix the 8
//    components of one channel) fuse into the GEMM epilogue.
//  * Weights are staged into LDS zero-padded to 32 out-channels -> the WMMA
//    inner loop is branch-free (one ds_load_b64 per B fragment; conflict-free
//    bank stride for CIN=28 since gcd(28,64)=4 gives 16 distinct banks for 16 lanes; CIN=56 gives 8).
//  * Geometric product + mv_layernorm is channel-local with a 28-wide channel
//    reduction -> one wave per row, Cayley/grade tables constexpr-generated
//    into __constant__, wave reduce via __shfl_xor (wave32).
//  * Activations (~63MB per edge stage) live in L2 (192MB); total DRAM traffic
//    ~1-2GB -> ~0.1ms at 23.3TB/s; compute goes through the WMMA pipe.
//  * Workspace: ~210MB of d_ws (3 edge ping-pong buffers + node buffers).

#include <hip/hip_runtime.h>
#include <cstdint>

typedef float v2f __attribute__((ext_vector_type(2)));
typedef float v8f __attribute__((ext_vector_type(8)));

#define DEV_EPS 1e-6f
#define RSQRT2  0.70710678118654752f

constexpr int BB   = 128;
constexpr int NNODE= 24;
constexpr int BN   = BB * NNODE;     // 3072
constexpr int EE   = 552;
constexpr int ETOT = BB * EE;        // 70656 (divisible by 16)
constexpr int CH   = 28;             // HID
constexpr int CV   = CH * 8;         // 224
constexpr int CH2  = 56;
constexpr int CV2  = CH2 * 8;        // 448

// ---------------- constexpr Cayley / grade tables ----------------
struct Tables {
  int   gr[8];          // grade of component
  int   res[8][8];      // res[a][b] = component index of mask(a)^mask(b)
  float sgn[8][8];      // Cayley sign
  int   pidx[4][4][4];  // grade-path -> gw column (lexicographic np.nonzero order)
};

constexpr int popc_(int x){ int c=0; while(x){ c+=x&1; x>>=1; } return c; }

constexpr Tables make_tables(){
  Tables t{};
  const int masks[8] = {0,1,2,4,3,5,6,7};
  int m2i[8] = {};
  for (int i=0;i<8;++i) m2i[masks[i]] = i;
  for (int i=0;i<8;++i) t.gr[i] = popc_(masks[i]);
  bool paths[4][4][4] = {};
  for (int a=0;a<8;++a)
    for (int b=0;b<8;++b){
      const int mc = masks[a]^masks[b];
      const int c  = m2i[mc];
      t.res[a][b] = c;
      int s=0, tt=masks[a]>>1;
      while (tt){ s += popc_(tt & masks[b]); tt >>= 1; }
      t.sgn[a][b] = (s&1) ? -1.f : 1.f;
      paths[t.gr[a]][t.gr[b]][t.gr[c]] = true;
    }
  int p=0;
  for (int i=0;i<4;++i)
    for (int j=0;j<4;++j)
      for (int k=0;k<4;++k)
        t.pidx[i][j][k] = paths[i][j][k] ? p++ : -1;   // p ends at N_PATHS==20
  return t;
}
__constant__ Tables dT = make_tables();

__device__ inline float sigm_(float z){ return 1.f/(1.f+__expf(-z)); }

__device__ inline float quad1_(const float* m9, const float* v){ // v^T M v over comps 1..3
  float q = 0.f;
  #pragma unroll
  for (int i=0;i<3;++i)
    #pragma unroll
    for (int j=0;j<3;++j) q += v[1+i]*m9[i*3+j]*v[1+j];
  return q;
}

// ---------------- setup: M = metric+metric^T, 3x3 Jacobi eigh, P, P^-1 ------
__global__ void k_setup(const float* __restrict__ metric, float* __restrict__ hdr){
  if (threadIdx.x != 0 || blockIdx.x != 0) return;
  float A[3][3], V[3][3];
  for (int i=0;i<3;++i)
    for (int j=0;j<3;++j){
      const float m = metric[i*3+j] + metric[j*3+i];
      A[i][j] = m; hdr[i*3+j] = m;
      V[i][j] = (i==j) ? 1.f : 0.f;
    }
  for (int sweep=0; sweep<24; ++sweep){
    const int PQ[3][2] = {{0,1},{0,2},{1,2}};
    for (int pp=0; pp<3; ++pp){
      const int p = PQ[pp][0], q = PQ[pp][1];
      const float apq = A[p][q];
      if (fabsf(apq) < 1e-14f) continue;
      const float theta = (A[q][q]-A[p][p])/(2.f*apq);
      const float tt = ((theta>=0.f)?1.f:-1.f)/(fabsf(theta)+sqrtf(theta*theta+1.f));
      const float c = 1.f/sqrtf(tt*tt+1.f), s = tt*c;
      for (int k=0;k<3;++k){ float akp=A[k][p], akq=A[k][q]; A[k][p]=c*akp-s*akq; A[k][q]=s*akp+c*akq; }
      for (int k=0;k<3;++k){ float apk=A[p][k], aqk=A[q][k]; A[p][k]=c*apk-s*aqk; A[q][k]=s*apk+c*aqk; }
      for (int k=0;k<3;++k){ float vkp=V[k][p], vkq=V[k][q]; V[k][p]=c*vkp-s*vkq; V[k][q]=s*vkp+c*vkq; }
    }
  }
  float ev[3] = {A[0][0], A[1][1], A[2][2]};
  int perm[3] = {0,1,2};
  for (int a=0;a<3;++a)
    for (int b=a+1;b<3;++b)
      if (ev[perm[b]] < ev[perm[a]]){ int t=perm[a]; perm[a]=perm[b]; perm[b]=t; }
  float P[3][3];
  for (int i=0;i<3;++i)
    for (int j=0;j<3;++j){ P[i][j] = V[i][perm[j]]; hdr[9+i*3+j] = P[i][j]; }
  const float det = P[0][0]*(P[1][1]*P[2][2]-P[1][2]*P[2][1])
                  - P[0][1]*(P[1][0]*P[2][2]-P[1][2]*P[2][0])
                  + P[0][2]*(P[1][0]*P[2][1]-P[1][1]*P[2][0]);
  const float id = 1.f/det;
  float inv[3][3];
  inv[0][0]= (P[1][1]*P[2][2]-P[1][2]*P[2][1])*id;
  inv[0][1]=-(P[0][1]*P[2][2]-P[0][2]*P[2][1])*id;
  inv[0][2]= (P[0][1]*P[1][2]-P[0][2]*P[1][1])*id;
  inv[1][0]=-(P[1][0]*P[2][2]-P[1][2]*P[2][0])*id;
  inv[1][1]= (P[0][0]*P[2][2]-P[0][2]*P[2][0])*id;
  inv[1][2]=-(P[0][0]*P[1][2]-P[0][2]*P[1][0])*id;
  inv[2][0]= (P[1][0]*P[2][1]-P[1][1]*P[2][0])*id;
  inv[2][1]=-(P[0][0]*P[2][1]-P[0][1]*P[2][0])*id;
  inv[2][2]= (P[0][0]*P[1][1]-P[0][1]*P[1][0])*id;
  for (int i=0;i<3;++i)
    for (int j=0;j<3;++j) hdr[18+i*3+j] = inv[i][j];
}

// ---------------- per-batch loc mean ----------------
__global__ void k_mean(const float* __restrict__ loc, float* __restrict__ mean){
  const int b = blockIdx.x, t = threadIdx.x;
  if (t < 3){
    float s = 0.f;
    for (int n=0;n<NNODE;++n) s += loc[((size_t)b*NNODE+n)*3 + t];
    mean[b*3+t] = s * (1.f/NNODE);
  }
}

// ---------------- rotate loc_mean and vel by P ----------------
__global__ void k_rot(const float* __restrict__ loc, const float* __restrict__ vel,
                      const float* __restrict__ mean, const float* __restrict__ hdr,
                      float* __restrict__ lmr, float* __restrict__ velr){
  const int bn = blockIdx.x*blockDim.x + threadIdx.x;
  if (bn >= BN) return;
  const int b = bn / NNODE;
  const float* P = hdr + 9;
  float lm[3], vv[3];
  for (int k=0;k<3;++k){ lm[k] = loc[(size_t)bn*3+k] - mean[b*3+k]; vv[k] = vel[(size_t)bn*3+k]; }
  for (int d=0;d<3;++d){
    float a=0.f, c=0.f;
    for (int k=0;k<3;++k){ a += lm[k]*P[k*3+d]; c += vv[k]*P[k*3+d]; }
    lmr[(size_t)bn*3+d]  = a;
    velr[(size_t)bn*3+d] = c;
  }
}

// ---------------- embedding: h = mv_linear_dense([inv,cov], emb_w, emb_b) ---
__global__ void k_embed(const float* __restrict__ charges, const float* __restrict__ lmr,
                        const float* __restrict__ velr, const float* __restrict__ embw,
                        const float* __restrict__ embb, float* __restrict__ h){
  const int id = blockIdx.x*blockDim.x + threadIdx.x;
  if (id >= BN*CH) return;
  const int bn = id / CH, n = id % CH;
  const float ch = charges[bn];
  const float w0 = embw[n*3+0], w1 = embw[n*3+1], w2 = embw[n*3+2];
  float* hp = h + (size_t)bn*CV + n*8;
  hp[0] = w0*ch + embb[n];
  for (int d=0;d<3;++d) hp[1+d] = w1*lmr[(size_t)bn*3+d] + w2*velr[(size_t)bn*3+d];
  hp[4]=0.f; hp[5]=0.f; hp[6]=0.f; hp[7]=0.f;
}

// ---------------- edge features: x_e = h[rows] - h[cols] ----------------
__global__ void k_ediff(const float* __restrict__ h, const int* __restrict__ edges,
                        float* __restrict__ xe){
  const long id = (long)blockIdx.x*blockDim.x + threadIdx.x;
  if (id >= (long)ETOT*CV) return;
  const long e = id / CV; const int t = (int)(id % CV);
  const int b = (int)(e / EE), j = (int)(e % EE);
  const int r = edges[(b*2+0)*EE + j] + b*NNODE;
  const int c = edges[(b*2+1)*EE + j] + b*NNODE;
  xe[id] = h[(size_t)r*CV + t] - h[(size_t)c*CV + t];
}

// ============ fused WMMA mv_linear (+bias / +silu / norm epilogue) ==========
// Workgroup: 8 wave32; wave i computes component i; 16 rows x 28 out-ch tile.
// Weights staged into LDS zero-padded to 32 out-ch, [grade][n][K] layout:
// the WMMA K-loop is branch-free and B fragments are single ds_load_b64s.
// EPI: 0 = +bias (wl+bl), 1 = +bias then mv_silu, 2 = norm_layer (no bias).
template<int CIN, int EPI>
__global__ __launch_bounds__(256) void k_mvlin(
    const float* __restrict__ x, const float* __restrict__ w,
    const float* __restrict__ bias, const float* __restrict__ aA,
    const float* __restrict__ aB, const float* __restrict__ hdr,
    float* __restrict__ y)
{
  const int comp = threadIdx.x >> 5;
  const int lane = threadIdx.x & 31;
  const int r    = lane & 15;            // A: row, B: out-channel within tile
  const int kh   = (lane >> 4) << 1;     // K sub-offset per lane half
  const long rowbase = (long)blockIdx.x * 16;
  const int g = dT.gr[comp];             // grade -> weight slice

  // ---- stage grade-sliced, zero-padded weights into LDS: smw[g][n(32)][CIN]
  __shared__ float smw[4 * 32 * CIN];
  for (int idx = threadIdx.x; idx < 4*32*CIN; idx += 256){
    const int g2  = idx / (32*CIN);
    const int rem = idx % (32*CIN);
    const int n   = rem / CIN;
    const int k   = rem % CIN;
    smw[idx] = (n < CH) ? w[(n*CIN + k)*4 + g2] : 0.f;
  }
  __syncthreads();
  const float* wg = smw + g * 32 * CIN;

  v8f acc0 = {0.f,0.f,0.f,0.f,0.f,0.f,0.f,0.f};
  v8f acc1 = acc0;

  #pragma unroll
  for (int k0 = 0; k0 < CIN; k0 += 4){
    const int kb = k0 + kh;              // even -> 8B-aligned LDS reads
    // A fragment 16x4 (rows x K) for component `comp`
    const float* xp = x + ((rowbase + r)*CIN + kb)*8 + comp;
    v2f a; a.x = xp[0]; a.y = xp[8];
    // B fragments 4x16 (K x out-channels) from LDS (zero-padded cols 28..31)
    const v2f b0 = *(const v2f*)&wg[ r       *CIN + kb];
    const v2f b1 = *(const v2f*)&wg[(16 + r)*CIN + kb];
    acc0 = __builtin_amdgcn_wmma_f32_16x16x4_f32(false, a, false, b0, (short)0, acc0, false, false);
    acc1 = __builtin_amdgcn_wmma_f32_16x16x4_f32(false, a, false, b1, (short)0, acc1, false, false);
  }

  // exchange the 8 components through LDS for the per-channel epilogue
  __shared__ float sm[16][32][8];
  #pragma unroll
  for (int v=0; v<8; ++v){
    const int rr = (lane < 16) ? v : v + 8;   // C layout: vgpr v -> M=v / M=v+8
    sm[rr][r     ][comp] = acc0[v];
    sm[rr][r + 16][comp] = acc1[v];
  }
  __syncthreads();

  float m9[9];
  if (EPI != 0){
    #pragma unroll
    for (int t=0;t<9;++t) m9[t] = hdr[t];
  }

  for (int idx = threadIdx.x; idx < 16*CH; idx += 256){
    const int row = idx & 15, col = idx >> 4;    // col < 28
    float v[8];
    #pragma unroll
    for (int i=0;i<8;++i) v[i] = sm[row][col][i];
    if (EPI != 2) v[0] += bias[col];             // b * E0
    if (EPI == 1){                               // mv_silu
      const float q1 = quad1_(m9, v);
      const float q2 = v[4]*v[4]+v[5]*v[5]+v[6]*v[6];
      const float q3 = v[7]*v[7];
      const float f[4] = {v[0], q1, q2, q3};
      float gg[4];
      #pragma unroll
      for (int s=0;s<4;++s) gg[s] = sigm_(aA[col*4+s]*f[s] + aB[col*4+s]);
      #pragma unroll
      for (int i=0;i<8;++i) v[i] *= gg[dT.gr[i]];
    }
    if (EPI == 2){                               // norm_layer
      const float q0 = v[0]*v[0];
      const float q1 = quad1_(m9, v);
      const float q2 = v[4]*v[4]+v[5]*v[5]+v[6]*v[6];
      const float q3 = v[7]*v[7];
      const float q[4] = {q0,q1,q2,q3};
      float nn[4];
      #pragma unroll
      for (int s=0;s<4;++s){
        const float n0 = sqrtf(fabsf(q[s]) + DEV_EPS);
        nn[s] = sigm_(aA[col*4+s])*(n0 - 1.f) + 1.f;
      }
      #pragma unroll
      for (int i=0;i<8;++i) v[i] /= (nn[dT.gr[i]] + DEV_EPS);
    }
    float* yp = y + ((rowbase + row)*CH + col)*8;
    #pragma unroll
    for (int i=0;i<8;++i) yp[i] = v[i];
  }
}

// ============ geometric product + residual combine + mv_layernorm ===========
// One wave32 per row; lane c = channel (c<28 live). In-place on y1 is safe:
// each lane only rewrites its own channel after reading it.
__global__ __launch_bounds__(256) void k_geo_ln(
    const float* __restrict__ y1, const float* __restrict__ xr,
    const float* __restrict__ y2, const float* __restrict__ gw,
    const float* __restrict__ lna, const float* __restrict__ hdr,
    float* __restrict__ out, long rows)
{
  const int wv = threadIdx.x >> 5, lane = threadIdx.x & 31;
  const long row = (long)blockIdx.x*8 + wv;
  const bool act = (lane < CH) && (row < rows);
  const long base = (row*CH + (act ? lane : 0))*8;

  float x8[8], xr8[8], y28[8], g20[20];
  #pragma unroll
  for (int i=0;i<8;++i){
    x8[i]  = act ? y1[base+i] : 0.f;
    xr8[i] = act ? xr[base+i] : 0.f;
    y28[i] = act ? y2[base+i] : 0.f;
  }
  #pragma unroll
  for (int p=0;p<20;++p) g20[p] = act ? gw[lane*20+p] : 0.f;

  // out_b = sum_a sgn(a,b) * w8[g(a),g(b),g(c)] * x_a * xr_c,  c = res(a,b)
  float o8[8] = {0.f,0.f,0.f,0.f,0.f,0.f,0.f,0.f};
  #pragma unroll
  for (int a=0;a<8;++a)
    #pragma unroll
    for (int b=0;b<8;++b){
      const int c  = dT.res[a][b];
      const int pi = dT.pidx[dT.gr[a]][dT.gr[b]][dT.gr[c]];
      o8[b] += dT.sgn[a][b] * g20[pi] * x8[a] * xr8[c];
    }

  float val[8];
  #pragma unroll
  for (int i=0;i<8;++i) val[i] = (y28[i] + o8[i]) * RSQRT2;

  float m9[9];
  #pragma unroll
  for (int t=0;t<9;++t) m9[t] = hdr[t];
  const float q0 = val[0]*val[0];
  const float q1 = quad1_(m9, val);
  const float q2 = val[4]*val[4]+val[5]*val[5]+val[6]*val[6];
  const float q3 = val[7]*val[7];
  float nch = act ? sqrtf(fabsf(q0+q1+q2+q3) + DEV_EPS) : 0.f;

  float tot = nch;                         // mean over 28 channels (wave reduce)
  for (int off=16; off; off >>= 1) tot += __shfl_xor(tot, off, 32);
  const float nm = tot * (1.f/28.f);

  if (act){
    const float s = lna[lane] / (nm + DEV_EPS);
    #pragma unroll
    for (int i=0;i<8;++i) out[base+i] = val[i]*s;
  }
}

// ---------------- segment sum (mean done at concat) ----------------
__global__ void k_agg(const float* __restrict__ msg, const int* __restrict__ edges,
                      float* __restrict__ agg, float* __restrict__ cnt){
  const long id = (long)blockIdx.x*blockDim.x + threadIdx.x;
  if (id >= (long)ETOT*CV) return;
  const long e = id / CV; const int t = (int)(id % CV);
  const int b = (int)(e / EE), j = (int)(e % EE);
  const int r = edges[(b*2+0)*EE + j] + b*NNODE;
  atomicAdd(&agg[(size_t)r*CV + t], msg[id]);
  if (t == 0) atomicAdd(&cnt[r], 1.f);
}

// ---------------- node input: xc = concat([h, agg/max(cnt,1)], ch) ----------
__global__ void k_concat(const float* __restrict__ h, const float* __restrict__ agg,
                         const float* __restrict__ cnt, float* __restrict__ xc){
  const long id = (long)blockIdx.x*blockDim.x + threadIdx.x;
  if (id >= (long)BN*CV2) return;
  const int bn = (int)(id / CV2), t = (int)(id % CV2);
  if (t < CV) xc[id] = h[(size_t)bn*CV + t];
  else        xc[id] = agg[(size_t)bn*CV + (t - CV)] / fmaxf(cnt[bn], 1.f);
}

// ---------------- residual: h += node_out ----------------
__global__ void k_resid(float* __restrict__ h, const float* __restrict__ d){
  const long id = (long)blockIdx.x*blockDim.x + threadIdx.x;
  if (id >= (long)BN*CV) return;
  h[id] += d[id];
}

// ---------------- projection + un-rotation ----------------
__global__ void k_proj(const float* __restrict__ h, const float* __restrict__ velr,
                       const float* __restrict__ projw, const float* __restrict__ hdr,
                       float* __restrict__ out){
  const int bn = blockIdx.x*blockDim.x + threadIdx.x;
  if (bn >= BN) return;
  const float* hp = h + (size_t)bn*CV;
  float pr[2][3];
  for (int n=0;n<2;++n)
    for (int d=0;d<3;++d){
      float s = 0.f;
      for (int m=0;m<CH;++m) s += hp[m*8+1+d] * projw[(n*CH+m)*4 + 1]; // grade-1 slice
      pr[n][d] = s;
    }
  const float* Pi = hdr + 18;
  for (int d=0;d<3;++d){
    float a=0.f, c=0.f;
    for (int k=0;k<3;++k){
      a += pr[0][k]*Pi[k*3+d];
      c += (velr[(size_t)bn*3+k] + pr[1][k])*Pi[k*3+d];
    }
    out[(size_t)bn*6 + d]     = a;   // loc_abs - loc@Pi == pred0_vec @ Pi
    out[(size_t)bn*6 + 3 + d] = c;   // vel_abs
  }
}

// ---------------- host-side plumbing ----------------
struct Blk { const float *bl,*gw,*lin_b,*lin_w,*ln_a,*na,*silu_a,*silu_b,*wl,*wr; };

static inline unsigned cdivu(long a, long b){ return (unsigned)((a + b - 1)/b); }

template<int CIN>
static void run_block(const Blk& p, const float* xin, float* y1, float* xrb,
                      float* y2b, long rows, const float* hdr, hipStream_t s){
  const dim3 grid((unsigned)(rows/16));
  // lin + bias + mv_silu
  k_mvlin<CIN,1><<<grid,256,0,s>>>(xin, p.lin_w, p.lin_b, p.silu_a, p.silu_b, hdr, y1);
  // wr + norm_layer
  k_mvlin<CH,2><<<grid,256,0,s>>>(y1, p.wr, nullptr, p.na, nullptr, hdr, xrb);
  // wl + bias
  k_mvlin<CH,0><<<grid,256,0,s>>>(y1, p.wl, p.bl, nullptr, nullptr, hdr, y2b);
  // geo + combine + mv_layernorm (writes result into y1)
  k_geo_ln<<<(unsigned)(rows/8),256,0,s>>>(y1, xrb, y2b, p.gw, p.ln_a, hdr, y1, rows);
}

extern "C" void kernel_launch(void* const* d_in, const int* in_sizes, int n_in,
                              void* d_out, int out_size, void* d_ws, size_t ws_size,
                              hipStream_t stream){
  (void)in_sizes; (void)n_in; (void)out_size; (void)ws_size;
  auto F = [&](int i){ return (const float*)d_in[i]; };

  // setup_inputs() order, then params flattened in JAX pytree (sorted-key) order.
  const float* loc     = F(0);
  const float* vel     = F(1);
  const float* charges = F(2);
  const int*   edges   = (const int*)d_in[3];
  const float* metric  = F(4);
  const float* emb_b   = F(5);
  const float* emb_w   = F(6);
  Blk blk[3][4];  // [layer][edge0, edge1, node0, node1]
  for (int li=0; li<3; ++li){
    const int base = 7 + 40*li;
    for (int q=0; q<4; ++q){
      const int o = base + q*10;  // keys: bl,gw,lin_b,lin_w,ln_a,na,silu_a,silu_b,wl,wr
      blk[li][q] = { F(o+0),F(o+1),F(o+2),F(o+3),F(o+4),F(o+5),F(o+6),F(o+7),F(o+8),F(o+9) };
    }
  }
  const float* proj_w = F(128);   // 127 = proj_b (comp0 only; unused for vec output)

  // workspace carve-up (floats); ~210MB total
  float* W = (float*)d_ws;
  size_t off = 0;
  float* HDR  = W + off; off += 64;                    // M(9) P(9) Pinv(9)
  float* MEAN = W + off; off += (size_t)BB*3;
  float* LMR  = W + off; off += (size_t)BN*3;
  float* VELR = W + off; off += (size_t)BN*3;
  float* H    = W + off; off += (size_t)BN*CV;
  float* N0   = W + off; off += (size_t)BN*CV;
  float* N1   = W + off; off += (size_t)BN*CV;
  float* N2   = W + off; off += (size_t)BN*CV;
  float* AGG  = W + off; off += (size_t)BN*CV;
  float* CNT  = W + off; off += (size_t)BN;
  float* XC   = W + off; off += (size_t)BN*CV2;
  float* E0b  = W + off; off += (size_t)ETOT*CV;
  float* E1b  = W + off; off += (size_t)ETOT*CV;
  float* E2b  = W + off; off += (size_t)ETOT*CV;

  k_setup<<<1,32,0,stream>>>(metric, HDR);
  k_mean <<<BB,32,0,stream>>>(loc, MEAN);
  k_rot  <<<cdivu(BN,256),256,0,stream>>>(loc, vel, MEAN, HDR, LMR, VELR);
  k_embed<<<cdivu((long)BN*CH,256),256,0,stream>>>(charges, LMR, VELR, emb_w, emb_b, H);

  for (int li=0; li<3; ++li){
    // ---- edge CEMLP on h[rows]-h[cols] ----
    k_ediff<<<cdivu((long)ETOT*CV,256),256,0,stream>>>(H, edges, E0b);
    run_block<CH>(blk[li][0], E0b, E1b, E2b, E0b, ETOT, HDR, stream); // out -> E1b
    run_block<CH>(blk[li][1], E1b, E0b, E2b, E1b, ETOT, HDR, stream); // out -> E0b (msg)
    // ---- segment mean ----
    hipMemsetAsync(AGG, 0, (size_t)BN*CV*sizeof(float), stream);
    hipMemsetAsync(CNT, 0, (size_t)BN*sizeof(float), stream);
    k_agg<<<cdivu((long)ETOT*CV,256),256,0,stream>>>(E0b, edges, AGG, CNT);
    k_concat<<<cdivu((long)BN*CV2,256),256,0,stream>>>(H, AGG, CNT, XC);
    // ---- node CEMLP on concat([h, agg]) ----
    run_block<CH2>(blk[li][2], XC, N0, N1, N2, BN, HDR, stream);      // out -> N0
    run_block<CH >(blk[li][3], N0, N1, N2, N0, BN, HDR, stream);      // out -> N1
    k_resid<<<cdivu((long)BN*CV,256),256,0,stream>>>(H, N1);
  }

  k_proj<<<cdivu(BN,256),256,0,stream>>>(H, VELR, proj_w, HDR, (float*)d_out);
}